// ProposalLayer_90486370992178
// MI455X (gfx1250) — compile-verified
//
#include <hip/hip_runtime.h>
#include <stdint.h>

// ---------------- static problem geometry ----------------
#define B_       16
#define A_       9
#define K_       32768            // H*W = 128*256
#define N_       (A_*K_)          // 294912 anchors per image (a-major: m = a*K + k)
#define PRE_     6000
#define POST_    300
#define CHUNK_   4096
#define NCH_     (N_/CHUNK_)      // 72
#define CSTRIDE_ 30000            // per-batch candidate region: 24000 box floats + 6000 scores
#define NEG_     (-1.0e30f)

// anchor widths/heights from py-faster-rcnn generator (base16, ratios .5/1/2, scales 8/16/32)
__constant__ float c_aw[9] = {184.f,368.f,736.f,128.f,256.f,512.f, 88.f,176.f,352.f};
__constant__ float c_ah[9] = { 96.f,192.f,384.f,128.f,256.f,512.f,176.f,352.f,704.f};

// ---------------- helpers ----------------
__device__ __forceinline__ uint32_t f2key(float f) {
  uint32_t u = __float_as_uint(f);
  return (u & 0x80000000u) ? ~u : (u | 0x80000000u);   // order-preserving float->uint
}
__device__ __forceinline__ float key2f(uint32_t k) {
  uint32_t u = (k & 0x80000000u) ? (k & 0x7FFFFFFFu) : ~k;
  return __uint_as_float(u);
}

__device__ __forceinline__ void decode_box(const float* __restrict__ lab,
                                           const float* __restrict__ reg,
                                           int b, int a, int k,
                                           float& score, float4& box, bool& keep) {
  size_t lb = (size_t)b * 18 * K_;
  size_t rb = (size_t)b * 36 * K_;
  float s  = lab[lb + (size_t)(2*a+1) * K_ + k];      // fg channel
  float dx = reg[rb + (size_t)(4*a+0) * K_ + k];
  float dy = reg[rb + (size_t)(4*a+1) * K_ + k];
  float dw = reg[rb + (size_t)(4*a+2) * K_ + k];
  float dh = reg[rb + (size_t)(4*a+3) * K_ + k];
  float aw = c_aw[a], ah = c_ah[a];
  int x = k & 255, y = k >> 8;                        // W=256
  float acx = (float)(x * 16) + 7.5f;                 // anchor center = shift + 7.5 for all anchors
  float acy = (float)(y * 16) + 7.5f;
  float pcx = dx * aw + acx;
  float pcy = dy * ah + acy;
  float pw  = expf(dw) * aw;
  float ph  = expf(dh) * ah;
  float x1 = pcx - 0.5f * pw, x2 = pcx + 0.5f * pw;
  float y1 = pcy - 0.5f * ph, y2 = pcy + 0.5f * ph;
  x1 = fminf(fmaxf(x1, 0.f), 4095.f);
  x2 = fminf(fmaxf(x2, 0.f), 4095.f);
  y1 = fminf(fmaxf(y1, 0.f), 2047.f);
  y2 = fminf(fmaxf(y2, 0.f), 2047.f);
  keep = (x2 - x1 + 1.f >= 16.f) && (y2 - y1 + 1.f >= 16.f);
  box.x = x1; box.y = y1; box.z = x2; box.w = y2;
  score = s;
}

// ---------------- 1) decode + sortable key (bandwidth-bound streaming) ----------------
__global__ __launch_bounds__(256) void decode_keys_kernel(const float* __restrict__ lab,
                                                          const float* __restrict__ reg,
                                                          uint32_t* __restrict__ keys) {
  int tid = blockIdx.x * 256 + threadIdx.x;
  if (tid >= B_ * N_) return;
  int b = tid / N_;
  int m = tid - b * N_;
  int a = m >> 15;                 // K = 2^15
  int k = m & (K_ - 1);
  float s; float4 bx; bool keep;
  decode_box(lab, reg, b, a, k, s, bx, keep);
  float masked = keep ? s : NEG_;
  keys[(size_t)b * N_ + m] = f2key(masked);
}

// ---------------- 2) zero scratch ----------------
__global__ __launch_bounds__(256) void zero_u32_kernel(uint32_t* p, int n) {
  int i = blockIdx.x * 256 + threadIdx.x;
  if (i < n) p[i] = 0u;
}

// ---------------- 3) per-batch radix histograms (LDS-privatized, integer atomics) ----------------
__global__ __launch_bounds__(256) void hist_kernel(const uint32_t* __restrict__ keys,
                                                   uint32_t* __restrict__ hist,
                                                   const uint32_t* __restrict__ state,
                                                   int pass) {
  int b = blockIdx.y;
  const uint32_t* kk = keys + (size_t)b * N_;
  uint32_t* h = hist + b * 4096;
  const uint32_t* st = state + b * 16;
  __shared__ uint32_t lh[4096];
  for (int i = threadIdx.x; i < 4096; i += 256) lh[i] = 0u;
  __syncthreads();
  uint32_t pfx12 = st[0];
  uint32_t pfx24 = (st[0] << 12) | st[1];
  int stride = gridDim.x * 256;
  for (int i = blockIdx.x * 256 + threadIdx.x; i < N_; i += stride) {
    __builtin_prefetch((const void*)(kk + i + stride), 0, 0);   // -> global_prefetch_b8
    uint32_t key = kk[i];
    if (pass == 0) {
      atomicAdd(&lh[key >> 20], 1u);
    } else if (pass == 1) {
      if ((key >> 20) == pfx12) atomicAdd(&lh[(key >> 8) & 0xFFFu], 1u);
    } else {
      if ((key >> 8) == pfx24) atomicAdd(&lh[key & 0xFFu], 1u);
    }
  }
  __syncthreads();
  for (int i = threadIdx.x; i < 4096; i += 256)
    if (lh[i]) atomicAdd(&h[i], lh[i]);
}

// ---------------- 4) pick boundary bin (one WG per batch) ----------------
__global__ __launch_bounds__(256) void pick_kernel(const uint32_t* __restrict__ hist,
                                                   uint32_t* __restrict__ state,
                                                   int nbins, int pass) {
  int b = blockIdx.x, t = threadIdx.x;
  const uint32_t* h = hist + b * 4096;
  uint32_t* st = state + b * 16;
  int per = nbins / 256;
  __shared__ uint32_t tot[256];
  uint32_t sum = 0;
  for (int i = 0; i < per; ++i) sum += h[t * per + i];
  tot[t] = sum;
  __syncthreads();
  uint32_t suf = 0;                         // keys in bins strictly above this thread's range
  for (int u = t + 1; u < 256; ++u) suf += tot[u];
  uint32_t target = (uint32_t)PRE_ - st[3]; // st[3] = count strictly greater so far (0 before pass 0)
  uint32_t acc = suf;
  for (int v = t * per + per - 1; v >= t * per; --v) {
    uint32_t hv = h[v];
    if (acc < target && acc + hv >= target) {
      st[pass] = (uint32_t)v;
      st[3] += acc;                          // now = count of keys strictly greater than chosen path
      if (pass == 2) {
        uint32_t T = (st[0] << 20) | (st[1] << 8) | (uint32_t)v;
        st[4] = T;                           // exact threshold key
        st[5] = st[3];                       // c_gt
        st[6] = (uint32_t)PRE_ - st[3];      // ties needed at T
      }
      break;
    }
    acc += hv;
  }
}

// ---------------- 5) per-chunk predicate counts ----------------
__global__ __launch_bounds__(256) void chunk_count_kernel(const uint32_t* __restrict__ keys,
                                                          const uint32_t* __restrict__ state,
                                                          uint32_t* __restrict__ cgt,
                                                          uint32_t* __restrict__ ceq) {
  int b = blockIdx.y, c = blockIdx.x;
  uint32_t T = state[b * 16 + 4];
  __shared__ uint32_t sg, se;
  if (threadIdx.x == 0) { sg = 0u; se = 0u; }
  __syncthreads();
  uint32_t g = 0, e = 0;
  const uint32_t* kk = keys + (size_t)b * N_ + (size_t)c * CHUNK_;
  for (int i = threadIdx.x; i < CHUNK_; i += 256) {
    uint32_t key = kk[i];
    g += (key > T);
    e += (key == T);
  }
  atomicAdd(&sg, g);
  atomicAdd(&se, e);
  __syncthreads();
  if (threadIdx.x == 0) { cgt[b * NCH_ + c] = sg; ceq[b * NCH_ + c] = se; }
}

// ---------------- 6) exclusive scan of chunk counts (tiny, one lane per batch) ----------------
__global__ void chunk_scan_kernel(const uint32_t* __restrict__ cgt,
                                  const uint32_t* __restrict__ ceq,
                                  uint32_t* __restrict__ ogt,
                                  uint32_t* __restrict__ oeq) {
  int b = blockIdx.x;
  if (threadIdx.x == 0) {
    uint32_t g = 0, e = 0;
    for (int c = 0; c < NCH_; ++c) {
      ogt[b * NCH_ + c] = g; g += cgt[b * NCH_ + c];
      oeq[b * NCH_ + c] = e; e += ceq[b * NCH_ + c];
    }
  }
}

// ---------------- 7) deterministic compaction via wave32 ballot/popc ----------------
__global__ __launch_bounds__(256) void compact_kernel(const float* __restrict__ lab,
                                                      const float* __restrict__ reg,
                                                      const uint32_t* __restrict__ keys,
                                                      const uint32_t* __restrict__ state,
                                                      const uint32_t* __restrict__ ogt,
                                                      const uint32_t* __restrict__ oeq,
                                                      float* __restrict__ cand) {
  int b = blockIdx.y, c = blockIdx.x;
  uint32_t T    = state[b * 16 + 4];
  uint32_t c_gt = state[b * 16 + 5];
  uint32_t need = state[b * 16 + 6];
  __shared__ uint32_t wg[8], we[8];
  __shared__ uint32_t run_g, run_e;
  if (threadIdx.x == 0) { run_g = ogt[b * NCH_ + c]; run_e = oeq[b * NCH_ + c]; }
  __syncthreads();
  int wave = threadIdx.x >> 5, lane = threadIdx.x & 31;
  uint32_t lmask = (1u << lane) - 1u;
  float* cb = cand + (size_t)b * CSTRIDE_;
  for (int r = 0; r < CHUNK_ / 256; ++r) {
    int m = c * CHUNK_ + r * 256 + threadIdx.x;
    uint32_t key = keys[(size_t)b * N_ + m];
    bool pg = key > T;
    bool pe = (key == T);
    uint32_t bg = (uint32_t)__ballot(pg);   // wave32: low 32 bits
    uint32_t be = (uint32_t)__ballot(pe);
    if (lane == 0) { wg[wave] = __popc(bg); we[wave] = __popc(be); }
    __syncthreads();
    uint32_t offg = run_g, offe = run_e, totg = 0, tote = 0;
    for (int w = 0; w < 8; ++w) {
      uint32_t g = wg[w], e = we[w];
      if (w < wave) { offg += g; offe += e; }
      totg += g; tote += e;
    }
    if (pg || pe) {
      uint32_t dst; bool ok;
      if (pg) { dst = offg + __popc(bg & lmask); ok = true; }
      else {
        uint32_t er = offe + __popc(be & lmask);    // global tie rank (deterministic scan order)
        ok = (er < need);
        dst = c_gt + er;
      }
      if (ok) {
        int a = m >> 15, k = m & (K_ - 1);
        float s; float4 bx; bool keep;
        decode_box(lab, reg, b, a, k, s, bx, keep);
        ((float4*)cb)[dst] = bx;                    // boxes at [0, 24000)
        cb[4 * PRE_ + dst] = key2f(key);            // scores at [24000, 30000)
      }
    }
    __syncthreads();
    if (threadIdx.x == 0) { run_g += totg; run_e += tote; }
    __syncthreads();
  }
}

// ---------------- 8) greedy NMS, whole candidate set in LDS (TDM-staged) ----------------
#if defined(__HIP_DEVICE_COMPILE__)
#if __has_builtin(__builtin_amdgcn_tensor_load_to_lds) && __has_builtin(__builtin_amdgcn_s_wait_tensorcnt)
#define HAVE_TDM 1
#endif
#endif

__device__ __forceinline__ void block_argmax(float v, int i, int tid,
                                             float* rv, int* ri,
                                             float& bv, int& bi) {
  for (int off = 16; off > 0; off >>= 1) {        // wave32 shuffle reduction
    float ov = __shfl_xor(v, off, 32);
    int   oi = __shfl_xor(i, off, 32);
    if (ov > v || (ov == v && oi < i)) { v = ov; i = oi; }
  }
  if ((tid & 31) == 0) { rv[tid >> 5] = v; ri[tid >> 5] = i; }
  __syncthreads();
  if (tid < 32) {
    float v2 = (tid < 16) ? rv[tid] : -3.0e38f;
    int   i2 = (tid < 16) ? ri[tid] : 0x7FFFFFFF;
    for (int off = 16; off > 0; off >>= 1) {
      float ov = __shfl_xor(v2, off, 32);
      int   oi = __shfl_xor(i2, off, 32);
      if (ov > v2 || (ov == v2 && oi < i2)) { v2 = ov; i2 = oi; }
    }
    if (tid == 0) { rv[0] = v2; ri[0] = i2; }
  }
  __syncthreads();
  bv = rv[0]; bi = ri[0];
}

__global__ __launch_bounds__(512) void nms_kernel(const float* __restrict__ cand,
                                                  float* __restrict__ out) {
  int b = blockIdx.x;
  int tid = threadIdx.x;
  __shared__ __align__(16) float smem[CSTRIDE_];   // 120 KB: [boxes 24000][scores 6000]
  __shared__ float rval[16];
  __shared__ int   ridx[16];
  const float* src = cand + (size_t)b * CSTRIDE_;

#ifdef HAVE_TDM
  if (tid < 32) {                                  // wave 0 issues one TDM DMA for the whole tile
    typedef unsigned int u32x4 __attribute__((ext_vector_type(4)));
    typedef int          i32x8 __attribute__((ext_vector_type(8)));
    typedef int          i32x4 __attribute__((ext_vector_type(4)));
    uint64_t ga  = (uint64_t)(uintptr_t)src;
    uint32_t lds = (uint32_t)(uintptr_t)&smem[0];  // low 32 bits of flat addr = LDS offset
    u32x4 g0;
    g0[0] = 1u;                                                   // count=1 (user D#)
    g0[1] = lds;                                                  // lds_addr
    g0[2] = (uint32_t)ga;                                         // global_addr[31:0]
    g0[3] = (uint32_t)((ga >> 32) & 0x01FFFFFFu) | (2u << 30);    // global_addr[56:32] | type=2
    i32x8 g1;
    g1[0] = (int)(2u << 16);                                      // data_size=2 (4B)
    g1[1] = (int)(((uint32_t)CSTRIDE_ & 0xFFFFu) << 16);          // tensor_dim0[15:0]
    g1[2] = (int)(((uint32_t)CSTRIDE_ >> 16) | (1u << 16));       // tensor_dim0[31:16] | tensor_dim1=1
    g1[3] = (int)(((uint32_t)CSTRIDE_ & 0xFFFFu) << 16);          // tile_dim0 = 30000
    g1[4] = 1;                                                    // tile_dim1 = 1
    g1[5] = CSTRIDE_;                                             // tensor_dim0_stride[31:0]
    g1[6] = (int)(((uint32_t)CSTRIDE_ & 0xFFFFu) << 16);          // t0s[47:32]=0 | t1s[15:0]
    g1[7] = 0;
    i32x4 g2 = (i32x4){0, 0, 0, 0};
    i32x4 g3 = (i32x4){0, 0, 0, 0};
    i32x8 g4 = (i32x8){0, 0, 0, 0, 0, 0, 0, 0};                   // clang-23 extra group (unused)
    __builtin_amdgcn_tensor_load_to_lds(g0, g1, g2, g3, g4, 0);
    __builtin_amdgcn_s_wait_tensorcnt(0);
  }
  __syncthreads();
#else
  for (int i = tid; i < CSTRIDE_; i += 512) smem[i] = src[i];
  __syncthreads();
#endif

  float4* sbox  = (float4*)smem;
  float*  slive = smem + 4 * PRE_;

  // initial argmax
  float lm = -3.0e38f; int li = 0;
  for (int j = tid; j < PRE_; j += 512) {
    float s = slive[j];
    if (s > lm) { lm = s; li = j; }
  }
  float bv; int bi;
  block_argmax(lm, li, tid, rval, ridx, bv, bi);

  float* oboxes  = out;                 // (16,300,4)
  float* oscores = out + B_ * POST_ * 4;

  for (int it = 0; it < POST_; ++it) {
    int sel = bi; float sval = bv;
    if (tid == 0) {
      float4 bx = sbox[sel];
      float* ob = oboxes + ((size_t)b * POST_ + it) * 4;
      ob[0] = bx.x; ob[1] = bx.y; ob[2] = bx.z; ob[3] = bx.w;
      oscores[b * POST_ + it] = sval;
    }
    float4 bs = sbox[sel];
    float area_s = (bs.z - bs.x + 1.f) * (bs.w - bs.y + 1.f);
    lm = -3.0e38f; li = 0;
    for (int j = tid; j < PRE_; j += 512) {   // fused suppress + next argmax
      float s = slive[j];
      if (j == sel) { slive[j] = NEG_; s = NEG_; }
      else if (s > -1.0e29f) {
        float4 bj = sbox[j];
        float xx1 = fmaxf(bs.x, bj.x), yy1 = fmaxf(bs.y, bj.y);
        float xx2 = fminf(bs.z, bj.z), yy2 = fminf(bs.w, bj.w);
        float inter = fmaxf(xx2 - xx1 + 1.f, 0.f) * fmaxf(yy2 - yy1 + 1.f, 0.f);
        float area_j = (bj.z - bj.x + 1.f) * (bj.w - bj.y + 1.f);
        float iou = inter / (area_s + area_j - inter);
        if (iou > 0.7f) { slive[j] = NEG_; s = NEG_; }
      }
      if (s > lm || (s == lm && j < li)) { lm = s; li = j; }
    }
    block_argmax(lm, li, tid, rval, ridx, bv, bi);
  }
}

// ---------------- host: workspace layout + launch sequence ----------------
static constexpr size_t SZ_KEYS  = (size_t)B_ * N_ * 4;          // 18,874,368
static constexpr size_t OFF_HIST = SZ_KEYS;
static constexpr size_t SZ_HIST  = (size_t)B_ * 4096 * 4;        // 262,144
static constexpr size_t OFF_STATE= OFF_HIST + SZ_HIST;
static constexpr size_t SZ_STATE = (size_t)B_ * 16 * 4;
static constexpr size_t OFF_CGT  = OFF_STATE + SZ_STATE;
static constexpr size_t SZ_CH    = (size_t)B_ * NCH_ * 4;
static constexpr size_t OFF_CEQ  = OFF_CGT + SZ_CH;
static constexpr size_t OFF_OGT  = OFF_CEQ + SZ_CH;
static constexpr size_t OFF_OEQ  = OFF_OGT + SZ_CH;
static constexpr size_t OFF_CAND = ((OFF_OEQ + SZ_CH + 255) / 256) * 256;

extern "C" void kernel_launch(void* const* d_in, const int* in_sizes, int n_in,
                              void* d_out, int out_size, void* d_ws, size_t ws_size,
                              hipStream_t stream) {
  const float* lab = (const float*)d_in[0];   // (16,18,128,256) f32
  const float* reg = (const float*)d_in[1];   // (16,36,128,256) f32
  float* out = (float*)d_out;                 // 16*300*4 boxes ++ 16*300 scores
  char* ws = (char*)d_ws;

  uint32_t* keys  = (uint32_t*)(ws);
  uint32_t* hist  = (uint32_t*)(ws + OFF_HIST);
  uint32_t* state = (uint32_t*)(ws + OFF_STATE);
  uint32_t* cgt   = (uint32_t*)(ws + OFF_CGT);
  uint32_t* ceq   = (uint32_t*)(ws + OFF_CEQ);
  uint32_t* ogt   = (uint32_t*)(ws + OFF_OGT);
  uint32_t* oeq   = (uint32_t*)(ws + OFF_OEQ);
  float*    cand  = (float*)   (ws + OFF_CAND);

  int tot = B_ * N_;
  decode_keys_kernel<<<(tot + 255) / 256, 256, 0, stream>>>(lab, reg, keys);

  int hs_words = B_ * 4096 + B_ * 16;         // hist + state are contiguous
  zero_u32_kernel<<<(hs_words + 255) / 256, 256, 0, stream>>>(hist, hs_words);
  hist_kernel<<<dim3(64, B_), 256, 0, stream>>>(keys, hist, state, 0);
  pick_kernel<<<B_, 256, 0, stream>>>(hist, state, 4096, 0);

  zero_u32_kernel<<<(B_ * 4096 + 255) / 256, 256, 0, stream>>>(hist, B_ * 4096);
  hist_kernel<<<dim3(64, B_), 256, 0, stream>>>(keys, hist, state, 1);
  pick_kernel<<<B_, 256, 0, stream>>>(hist, state, 4096, 1);

  zero_u32_kernel<<<(B_ * 4096 + 255) / 256, 256, 0, stream>>>(hist, B_ * 4096);
  hist_kernel<<<dim3(64, B_), 256, 0, stream>>>(keys, hist, state, 2);
  pick_kernel<<<B_, 256, 0, stream>>>(hist, state, 256, 2);

  chunk_count_kernel<<<dim3(NCH_, B_), 256, 0, stream>>>(keys, state, cgt, ceq);
  chunk_scan_kernel<<<B_, 32, 0, stream>>>(cgt, ceq, ogt, oeq);
  compact_kernel<<<dim3(NCH_, B_), 256, 0, stream>>>(lab, reg, keys, state, ogt, oeq, cand);

  nms_kernel<<<B_, 512, 0, stream>>>(cand, out);
}